// HandshakingKernelSlow_68049461837834
// MI455X (gfx1250) — compile-verified
//
#include <hip/hip_runtime.h>
#include <hip/hip_bf16.h>
#include <math.h>

// Problem constants (reference: B=4, L=256, H=768)
#define Bsz 4
#define Lsz 256
#define Hsz 768
#define H2  1536
#define Psz 32896   // L*(L+1)/2

typedef __attribute__((ext_vector_type(2))) float v2f;
typedef __attribute__((ext_vector_type(4))) float v4f;
typedef __attribute__((ext_vector_type(8))) float v8f;

// ---------------------------------------------------------------------------
// Kernel 1: UV[m, n] = sum_k A[m,k] * Wcat[k,n]
//   A    = seq_hiddens flattened (B*L=1024, H=768), row-major
//   Wcat[k,n] = (n < H) ? W[n, k] : W[n-H, H+k]   (W row-major (H, 2H))
//   -> UV row m holds U (cols 0..767) and V (cols 768..1535) for token m.
// fp32 WMMA 16x16x4; one wave computes a 16x64 output strip (A-frag reused 4x).
// M tiles = 64, N groups of 64 = 24 -> 1536 waves -> 384 blocks of 4 waves.
// ---------------------------------------------------------------------------
__global__ __launch_bounds__(128) void handshake_gemm(
    const float* __restrict__ A, const float* __restrict__ W,
    float* __restrict__ UV)
{
    const int wave  = threadIdx.x >> 5;
    const int lane  = threadIdx.x & 31;
    const int gw    = blockIdx.x * 4 + wave;
    const int mt    = gw & 63;    // 0..63  : M tile (16 rows each)
    const int nt64  = gw >> 6;    // 0..23  : N group (64 cols each)
    const int row   = lane & 15;
    const int khalf = (lane >> 4) << 1;   // lanes 0-15 -> K{0,1}; 16-31 -> K{2,3}

    // A fragment pointer: per-lane float2 at (m, k+khalf)
    const float* aptr = A + (size_t)(mt * 16 + row) * Hsz + khalf;

    // B fragment pointers: Wcat is contiguous in k, so per-lane float2 loads.
    // A 64-wide N group lies entirely in one half (768 = 12 * 64).
    const bool half = (nt64 >= 12);
    const float* wptr[4];
#pragma unroll
    for (int nt = 0; nt < 4; ++nt) {
        int n  = nt64 * 64 + nt * 16 + row;
        int nl = half ? (n - Hsz) : n;
        wptr[nt] = W + (size_t)nl * H2 + (half ? Hsz : 0) + khalf;
    }

    v8f acc[4] = {};
#pragma unroll 4
    for (int k = 0; k < Hsz; k += 4) {
        v2f a = *(const v2f*)(aptr + k);
#pragma unroll
        for (int nt = 0; nt < 4; ++nt) {
            v2f bf = *(const v2f*)(wptr[nt] + k);
            // (neg_a, A, neg_b, B, c_mod, C, reuse_a, reuse_b)
            acc[nt] = __builtin_amdgcn_wmma_f32_16x16x4_f32(
                false, a, false, bf, (short)0, acc[nt], false, false);
        }
    }

    // C/D layout: VGPR r -> M = r + 8*(lane>=16), N = lane&15
    const int mbase = mt * 16 + ((lane >> 4) << 3);
#pragma unroll
    for (int nt = 0; nt < 4; ++nt) {
        const int ncol = nt64 * 64 + nt * 16 + row;
#pragma unroll
        for (int r = 0; r < 8; ++r) {
            UV[(size_t)(mbase + r) * H2 + ncol] = acc[nt][r];
        }
    }
}

// ---------------------------------------------------------------------------
// Kernel 2: out[b, p, h] = tanh(U[b, ii[p], h] + V[b, jj[p], h] + bias[h])
// One block per (p, b); 192 threads x float4 = 768 floats.
// Unrank p -> (i, j) for np.triu_indices order:
//   rev = P-1-p; t = row-from-end with T(t) <= rev < T(t+1);
//   i = L-1-t; j = L-1-(rev - T(t)).
// Writes are non-temporal (404 MB >> L2); UV reads stay cached (6 MB in L2).
// ---------------------------------------------------------------------------
__global__ __launch_bounds__(192) void handshake_combine(
    const float* __restrict__ UV, const float* __restrict__ bias,
    float* __restrict__ out)
{
    const int p   = blockIdx.x;
    const int b   = blockIdx.y;
    const int rev = (Psz - 1) - p;

    int t = (int)((sqrtf(8.0f * (float)rev + 1.0f) - 1.0f) * 0.5f);
    while ((t + 1) * (t + 2) / 2 <= rev) ++t;   // fix sqrt rounding
    while (t * (t + 1) / 2 > rev) --t;
    const int i = (Lsz - 1) - t;
    const int j = (Lsz - 1) - (rev - t * (t + 1) / 2);

    const int h = threadIdx.x << 2;   // 0,4,...,764
    const v4f u4 = *(const v4f*)(UV + (size_t)(b * Lsz + i) * H2 + h);
    const v4f v4 = *(const v4f*)(UV + (size_t)(b * Lsz + j) * H2 + Hsz + h);
    const v4f b4 = *(const v4f*)(bias + h);

    v4f o;
    o.x = tanhf(u4.x + v4.x + b4.x);
    o.y = tanhf(u4.y + v4.y + b4.y);
    o.z = tanhf(u4.z + v4.z + b4.z);
    o.w = tanhf(u4.w + v4.w + b4.w);

    v4f* dst = (v4f*)(out + (size_t)(b * Psz + p) * Hsz + h);
    __builtin_nontemporal_store(o, dst);
}

// ---------------------------------------------------------------------------
extern "C" void kernel_launch(void* const* d_in, const int* in_sizes, int n_in,
                              void* d_out, int out_size, void* d_ws, size_t ws_size,
                              hipStream_t stream)
{
    const float* seq  = (const float*)d_in[0];   // (B, L, H) fp32
    const float* W    = (const float*)d_in[1];   // (H, 2H)  fp32
    const float* bias = (const float*)d_in[2];   // (H,)     fp32
    float*       out  = (float*)d_out;           // (B, P, H) fp32
    float*       UV   = (float*)d_ws;            // needs 1024*1536*4 = 6 MB

    // GEMM: 1536 wave-tiles, 4 waves/block
    handshake_gemm<<<384, 128, 0, stream>>>(seq, W, UV);
    // Combine: one block per (p, b)
    handshake_combine<<<dim3(Psz, Bsz), 192, 0, stream>>>(UV, bias, out);
}